// MyUrsaGau_36773509989121
// MI455X (gfx1250) — compile-verified
//
#include <hip/hip_runtime.h>

// out[b, m] = sum_n exp(-0.5 * ||pts[b,n] - stars[m]||^2)
// B=32, N=4096, M=1024, D=3.
//
// Exponent formed by V_WMMA_F32_16X16X4_F32 (K=4):
//   A row (point n) = [x, y, z, ||x||^2]
//   B col (star m) = log2(e) * [sx, sy, sz, -0.5]          (loop-invariant)
//   C[.,m]         = -0.5 * log2(e) * ||s_m||^2            (loop-invariant)
//   D = log2(e) * (-0.5 * d2)  ->  term = exp2(D)  (single v_exp_f32 each)

typedef __attribute__((ext_vector_type(2))) float v2f;
typedef __attribute__((ext_vector_type(8))) float v8f;

#define B_DIM   32
#define N_PTS   4096
#define M_STARS 1024
#define LOG2E   1.44269504088896340736f

__global__ __launch_bounds__(256) void ursa_gau_wmma(
    const float* __restrict__ pts,    // (B, N, 3)
    const float* __restrict__ stars,  // (M, 3)
    float* __restrict__ out)          // (B, M)
{
    const int mtile = blockIdx.x;         // 0..63  -> m-tile of 16 stars
    const int b     = blockIdx.y;         // 0..31  -> batch
    const int tid   = threadIdx.x;        // 0..255 (8 wave32)
    const int wave  = tid >> 5;
    const int lane  = tid & 31;
    const int l16   = lane & 15;
    const bool hi   = (lane >= 16);       // upper half-wave holds K=2,3

    const int m_base = mtile * 16;

    // ---- loop-invariant B fragment (stars on columns) + C fragment ----
    // B layout (4x16 f32, 2 VGPRs): lane l, VGPR j -> B[K=(l/16)*2+j, n=l%16]
    const int   m  = m_base + l16;
    const float sx = stars[3 * m + 0];
    const float sy = stars[3 * m + 1];
    const float sz = stars[3 * m + 2];
    const float s2 = sx * sx + sy * sy + sz * sz;
    v2f bmat;
    bmat[0] = (hi ? sz : sx) * LOG2E;                  // K=2 : K=0
    bmat[1] = hi ? (-0.5f * LOG2E) : (sy * LOG2E);     // K=3 : K=1
    const float cval = -0.5f * LOG2E * s2;             // depends only on column m
    v8f cvec;
#pragma unroll
    for (int r = 0; r < 8; ++r) cvec[r] = cval;

    const float* ptsb = pts + (size_t)b * (N_PTS * 3);
    float acc = 0.0f;

    // 256 n-tiles of 16 points; 8 waves -> 32 tiles per wave (blocked).
    const int tiles_per_wave = (N_PTS / 16) / 8;
#pragma unroll 2
    for (int i = 0; i < tiles_per_wave; ++i) {
        const int n = (wave * tiles_per_wave + i) * 16 + l16;
        // A layout (16x4 f32, 2 VGPRs): lane l, VGPR j -> A[row=l%16, K=(l/16)*2+j]
        const float x  = ptsb[3 * n + 0];
        const float y  = ptsb[3 * n + 1];
        const float z  = ptsb[3 * n + 2];
        const float p2 = x * x + y * y + z * z;
        v2f amat;
        amat[0] = hi ? z  : x;     // K=2 : K=0
        amat[1] = hi ? p2 : y;     // K=3 : K=1

        // D[n_local, m_local] = -0.5*log2(e)*d2  (16x16 f32 in 8 VGPRs)
        v8f d = __builtin_amdgcn_wmma_f32_16x16x4_f32(
            /*neg_a=*/false, amat, /*neg_b=*/false, bmat,
            /*c_mod=*/(short)0, cvec, /*reuse_a=*/false, /*reuse_b=*/false);

        // Sum rows (r = n-dimension) in-lane: one v_exp_f32 per term.
        float s = 0.0f;
#pragma unroll
        for (int r = 0; r < 8; ++r) s += __builtin_amdgcn_exp2f(d[r]);
        acc += s;
    }

    // Lane l and lane l+16 hold rows 0-7 / 8-15 of each tile for the same m.
    float tot = acc + __shfl_xor(acc, 16, 32);

    // Combine the 8 waves' partial sums for this (b, m-tile).
    __shared__ float partial[8][16];
    if (lane < 16) partial[wave][l16] = tot;
    __syncthreads();
    if (tid < 16) {
        float s = 0.0f;
#pragma unroll
        for (int w = 0; w < 8; ++w) s += partial[w][tid];
        out[(size_t)b * M_STARS + m_base + tid] = s;
    }
}

extern "C" void kernel_launch(void* const* d_in, const int* in_sizes, int n_in,
                              void* d_out, int out_size, void* d_ws, size_t ws_size,
                              hipStream_t stream) {
    const float* pts   = (const float*)d_in[0];   // (32, 4096, 3) f32
    const float* stars = (const float*)d_in[1];   // (1024, 3) f32
    float* out         = (float*)d_out;           // (32, 1024) f32

    dim3 grid(M_STARS / 16, B_DIM);               // (64, 32) blocks, each owns (b, m-tile)
    ursa_gau_wmma<<<grid, 256, 0, stream>>>(pts, stars, out);
}